// ExiGCNLoRA_19782619365925
// MI455X (gfx1250) — compile-verified
//
#include <hip/hip_runtime.h>
#include <hip/hip_bf16.h>

// ---------------------------------------------------------------------------
// 3-layer GCN: z = spmm(spmm(spmm(X)@W1 relu)@W2 relu)@W3 + biases
// N = 100000 nodes, E = 1.6M edges, D = 128 (hidden), 64 (out)
//
// SpMM is the bottleneck (~5 GB of gather+atomic traffic); GEMMs total
// ~8.2 GFLOP and are done in exact f32 via V_WMMA_F32_16X16X4_F32.
// ---------------------------------------------------------------------------

typedef __attribute__((ext_vector_type(2))) float v2f;
typedef __attribute__((ext_vector_type(8))) float v8f;

// -------------------- SpMM: one wave32 per edge ----------------------------
// lane l handles columns [4l, 4l+3] of the 128-wide feature row.
__global__ __launch_bounds__(256) void gcn_spmm(
    const float* __restrict__ h,
    const int*   __restrict__ esrc,
    const int*   __restrict__ edst,
    const float* __restrict__ ew,
    float*       __restrict__ out,
    int E)
{
    const int wave = (int)((blockIdx.x * blockDim.x + threadIdx.x) >> 5);
    const int lane = threadIdx.x & 31;
    if (wave >= E) return;

    const int   s = esrc[wave];
    const int   d = edst[wave];
    const float w = ew[wave];

    const float4 v = reinterpret_cast<const float4*>(h + (size_t)s * 128)[lane];
    float* o = out + (size_t)d * 128 + lane * 4;
    unsafeAtomicAdd(o + 0, v.x * w);   // -> global_atomic_add_f32
    unsafeAtomicAdd(o + 1, v.y * w);
    unsafeAtomicAdd(o + 2, v.z * w);
    unsafeAtomicAdd(o + 3, v.w * w);
}

// -------------------- GEMM + bias (+ReLU) via f32 WMMA ---------------------
// Block = (NOUT/16) waves. Each block computes a 16 x NOUT output tile for
// rows [blockIdx.x*16, +16). A tile staged through LDS (shared by all waves);
// each wave owns one 16x16 column tile, K=128 swept in 32 wmma_f32_16x16x4.
//
// Fragment layouts (CDNA5 ISA 7.12.2, 32-bit operands, wave32):
//   A 16x4 : lane(0-15)=M, grp=lane>>4: VGPR0 holds K=2*grp, VGPR1 K=2*grp+1
//   B 4x16 : lane(0-15)=N (mirrored K split across lane halves)
//   C 16x16: VGPR r: lanes0-15 -> (M=r, N=lane); lanes16-31 -> (M=r+8)
template <int NOUT, bool RELU>
__global__ __launch_bounds__(NOUT * 2) void gcn_gemm(
    const float* __restrict__ A,     // [N, 128] aggregated features
    const float* __restrict__ W,     // [128, NOUT]
    const float* __restrict__ bias,  // [NOUT]
    float*       __restrict__ out,   // [N, NOUT]
    int N)
{
    constexpr int K   = 128;
    constexpr int LDA = K + 4;       // pad: stride 132 dwords -> conflict-free cols
    __shared__ float tileA[16 * LDA];

    const int       tid    = threadIdx.x;
    const long long m_base = (long long)blockIdx.x * 16;

    // Stage the 16x128 A tile (coalesced: consecutive tids -> consecutive cols).
    for (int idx = tid; idx < 16 * K; idx += blockDim.x) {
        const int r = idx >> 7;
        const int c = idx & (K - 1);
        const long long row = m_base + r;
        tileA[r * LDA + c] = (row < N) ? A[row * (long long)K + c] : 0.0f;
    }
    __syncthreads();

    const int wv   = tid >> 5;
    const int lane = tid & 31;
    const int l16  = lane & 15;
    const int g    = lane >> 4;      // 0: K+0/K+1, 1: K+2/K+3
    const int n    = wv * 16 + l16;  // output column owned by this lane

    v8f acc = {0.f, 0.f, 0.f, 0.f, 0.f, 0.f, 0.f, 0.f};

#pragma unroll
    for (int k0 = 0; k0 < K; k0 += 4) {
        const int ka = k0 + 2 * g;
        v2f a, b;
        a.x = tileA[l16 * LDA + ka];
        a.y = tileA[l16 * LDA + ka + 1];
        b.x = W[(size_t)ka * NOUT + n];
        b.y = W[(size_t)(ka + 1) * NOUT + n];
        // (neg_a, A, neg_b, B, c_mod, C, reuse_a, reuse_b)
        acc = __builtin_amdgcn_wmma_f32_16x16x4_f32(
            false, a, false, b, (short)0, acc, false, false);
    }

    const float bv = bias[n];
#pragma unroll
    for (int r = 0; r < 8; ++r) {
        const long long row = m_base + r + 8 * g;
        if (row < N) {
            float v = acc[r] + bv;
            if (RELU) v = fmaxf(v, 0.0f);
            out[row * (long long)NOUT + n] = v;
        }
    }
}

// ---------------------------------------------------------------------------
extern "C" void kernel_launch(void* const* d_in, const int* in_sizes, int n_in,
                              void* d_out, int out_size, void* d_ws, size_t ws_size,
                              hipStream_t stream)
{
    const float* features = (const float*)d_in[0];
    const int*   esrc     = (const int*)  d_in[1];
    const int*   edst     = (const int*)  d_in[2];
    const float* ew       = (const float*)d_in[3];
    const float* W1       = (const float*)d_in[4];
    const float* b1       = (const float*)d_in[5];
    const float* W2       = (const float*)d_in[6];
    const float* b2       = (const float*)d_in[7];
    const float* W3       = (const float*)d_in[8];
    const float* b3       = (const float*)d_in[9];

    const int N = in_sizes[0] / 128;
    const int E = in_sizes[1];

    float* agg  = (float*)d_ws;                 // [N,128] spmm accumulator
    float* hbuf = agg + (size_t)N * 128;        // [N,128] layer activations
    const size_t nbytes = (size_t)N * 128 * sizeof(float);

    const int eblocks = (E + 7) / 8;            // 8 waves (edges) per block
    const int mblocks = (N + 15) / 16;          // 16 output rows per block

    // Layer 1
    hipMemsetAsync(agg, 0, nbytes, stream);
    gcn_spmm<<<eblocks, 256, 0, stream>>>(features, esrc, edst, ew, agg, E);
    gcn_gemm<128, true><<<mblocks, 256, 0, stream>>>(agg, W1, b1, hbuf, N);

    // Layer 2
    hipMemsetAsync(agg, 0, nbytes, stream);
    gcn_spmm<<<eblocks, 256, 0, stream>>>(hbuf, esrc, edst, ew, agg, E);
    gcn_gemm<128, true><<<mblocks, 256, 0, stream>>>(agg, W2, b2, hbuf, N);

    // Layer 3 (no activation), straight to d_out [N, 64]
    hipMemsetAsync(agg, 0, nbytes, stream);
    gcn_spmm<<<eblocks, 256, 0, stream>>>(hbuf, esrc, edst, ew, agg, E);
    gcn_gemm<64, false><<<mblocks, 128, 0, stream>>>(agg, W3, b3, (float*)d_out, N);
}